// FeatureInteraction_83056077570560
// MI455X (gfx1250) — compile-verified
//
#include <hip/hip_runtime.h>

typedef __attribute__((ext_vector_type(2))) float v2f;
typedef __attribute__((ext_vector_type(4))) float v4f;
typedef __attribute__((ext_vector_type(8))) float v8f;

#define NF 27
#define FD 128
#define LDS_STRIDE 132   // 128 + 4 pad: kills LDS bank conflicts, keeps 16B row alignment
#define FLAT 3456        // 27*128
#define OUT_STRIDE 3807  // 3456 + 351

__global__ __launch_bounds__(128) void fi_wmma_kernel(const float* __restrict__ in,
                                                      float* __restrict__ out) {
    __shared__ float lds[32 * LDS_STRIDE];
    const int b   = blockIdx.x;
    const int tid = threadIdx.x;
    const float* __restrict__ src = in  + (size_t)b * FLAT;
    float*       __restrict__ dst = out + (size_t)b * OUT_STRIDE;

    // ---- Stage 27x128 fp32 tile into LDS (16B vector loads/stores) ----
    for (int idx = tid; idx < FLAT / 4; idx += 128) {
        const int row = idx >> 5;            // 32 float4 per row
        const int c4  = (idx & 31) << 2;
        v4f v = *(const v4f*)(src + (idx << 2));
        *(v4f*)(&lds[row * LDS_STRIDE + c4]) = v;
    }
    // Zero rows 27..31 so the padded 32x32 Gram tiles are well-defined.
    {
        v4f z = {};
        for (int idx = tid; idx < 5 * 32; idx += 128) {
            const int row = NF + (idx >> 5);
            const int c4  = (idx & 31) << 2;
            *(v4f*)(&lds[row * LDS_STRIDE + c4]) = z;
        }
    }
    __syncthreads();

    // ---- Flat embedding pass-through (coalesced b32 stores; out base only 4B aligned) ----
    for (int p = tid; p < FLAT; p += 128) {
        dst[p] = lds[(p >> 7) * LDS_STRIDE + (p & 127)];
    }

    // ---- Gram matrix via V_WMMA_F32_16X16X4_F32: one 16x16 tile per wave ----
    const int wave = tid >> 5;
    const int lane = tid & 31;
    const int half = lane >> 4;   // selects K pair {0,1} vs {2,3} within each step
    const int l16  = lane & 15;
    const int tr   = wave >> 1;   // tile row   (0..1)
    const int tc   = wave & 1;    // tile col   (0..1)

    const float* __restrict__ Arow = &lds[(tr * 16 + l16) * LDS_STRIDE];
    const float* __restrict__ Brow = &lds[(tc * 16 + l16) * LDS_STRIDE];

    v8f acc = {};
    #pragma unroll 8
    for (int k = 0; k < FD; k += 4) {
        const int kk = k + 2 * half;
        v2f a  = *(const v2f*)(Arow + kk);   // A: row = tr*16 + l16, cols kk,kk+1
        v2f bb = *(const v2f*)(Brow + kk);   // B = A^T: col = tc*16 + l16, rows kk,kk+1
        acc = __builtin_amdgcn_wmma_f32_16x16x4_f32(
            /*neg_a=*/false, a, /*neg_b=*/false, bb,
            /*c_mod=*/(short)0, acc, /*reuse_a=*/false, /*reuse_b=*/false);
    }

    // ---- Scatter upper triangle (i<j, j<27) into triu(27, k=1) row-major order ----
    const int j = tc * 16 + l16;
    #pragma unroll
    for (int v = 0; v < 8; ++v) {
        const int i = tr * 16 + v + 8 * half;   // C/D layout: VGPR v, lane half -> M = v + 8*half
        if (i < j && j < NF) {
            dst[FLAT + ((i * (53 - i)) >> 1) + (j - i - 1)] = acc[v];
        }
    }
}

extern "C" void kernel_launch(void* const* d_in, const int* in_sizes, int n_in,
                              void* d_out, int out_size, void* d_ws, size_t ws_size,
                              hipStream_t stream) {
    (void)n_in; (void)out_size; (void)d_ws; (void)ws_size;
    const float* emb = (const float*)d_in[0];
    float* out = (float*)d_out;
    const int batch = in_sizes[0] / FLAT;   // 16384
    fi_wmma_kernel<<<batch, 128, 0, stream>>>(emb, out);
}